// QCNNClassifier_71339406787263
// MI455X (gfx1250) — compile-verified
//
#include <hip/hip_runtime.h>
#include <stdint.h>

// ---------------------------------------------------------------------------
// QCNN 20-qubit statevector simulation for gfx1250 (MI455X).
//
// Strategy:
//  * Fuse each conv block + pool CX into one 4x4 complex unitary U (19 total),
//    realified into an 8x8 matrix W so that gate application is the GEMM
//    Y(16x8) = X(16x8) * W(8x8) over groups of 4 amplitudes.
//  * Two LDS-tiled passes over the 8MB state apply all 19 gates:
//      pass1: local index bits 0..11   (contiguous 32KB tiles, 10 gates)
//      pass2: local index bits {0..3,11..19} (gathered 64KB tiles, 9 gates)
//  * Tiles moved global<->LDS with the Tensor Data Mover
//    (tensor_load_to_lds / tensor_store_from_lds + s_wait_tensorcnt).
//  * Gate math on the matrix core: two chained V_WMMA_F32_16X16X4_F32 (K=8).
//  * Output = P(q0=0) = sum |amp|^2 over flat indices < 2^19  (== (<Z0>+1)/2).
// ---------------------------------------------------------------------------

typedef float v2f __attribute__((ext_vector_type(2)));
typedef float v8f __attribute__((ext_vector_type(8)));
typedef unsigned int v4u __attribute__((ext_vector_type(4)));
typedef int v8i __attribute__((ext_vector_type(8)));
typedef int v4i __attribute__((ext_vector_type(4)));

struct cplx { float re, im; };
__device__ __forceinline__ cplx cmul(cplx a, cplx b) {
    return { a.re * b.re - a.im * b.im, a.re * b.im + a.im * b.re };
}

// ---------------------------------------------------------------------------
// Init: zero state, set |0...0>, build 19 realified 8x8 gate matrices W.
// Gate t (qubit pair, params 3t..3t+2):
//   U = CX(q2->q1) . RZ_q2(th3) . CX(q1->q2) . (RZ_q1(th1) (x) RY_q2(th2))
// basis |b1 b2>, b1 = higher bit (q1), b2 = lower bit (q2).
// W[2k+s][2c+t] realifies y_c = sum_k U[c][k] x_k.
// ---------------------------------------------------------------------------
__global__ void qcnn_init(const float* __restrict__ params,
                          float* __restrict__ state,
                          float* __restrict__ Wtab) {
    const unsigned gid = blockIdx.x * blockDim.x + threadIdx.x;
    const unsigned nthr = gridDim.x * blockDim.x;
    for (unsigned i = gid; i < (2u << 20); i += nthr)
        state[i] = (i == 0u) ? 1.0f : 0.0f;

    if (gid < 19u) {
        const float t1 = 0.5f * params[3u * gid + 0u];
        const float t2 = 0.5f * params[3u * gid + 1u];
        const float t3 = 0.5f * params[3u * gid + 2u];
        const float c = cosf(t2), s = sinf(t2);
        cplx e0 = { cosf(t1), -sinf(t1) };          // exp(-i th1/2)
        cplx e1 = { e0.re, -e0.im };
        cplx f0 = { cosf(t3), -sinf(t3) };          // exp(-i th3/2)
        cplx f1 = { f0.re, -f0.im };
        cplx f0e0 = cmul(f0, e0), f1e1 = cmul(f1, e1);
        cplx f0e1 = cmul(f0, e1), f1e0 = cmul(f1, e0);
        cplx U[4][4] = {};
        U[0][0] = { f0e0.re * c,  f0e0.im * c };
        U[0][1] = { -f0e0.re * s, -f0e0.im * s };
        U[1][2] = { f1e1.re * c,  f1e1.im * c };
        U[1][3] = { -f1e1.re * s, -f1e1.im * s };
        U[2][2] = { f0e1.re * s,  f0e1.im * s };
        U[2][3] = { f0e1.re * c,  f0e1.im * c };
        U[3][0] = { f1e0.re * s,  f1e0.im * s };
        U[3][1] = { f1e0.re * c,  f1e0.im * c };
        float* Wg = Wtab + gid * 64u;
        for (int cc = 0; cc < 4; ++cc)
            for (int k = 0; k < 4; ++k) {
                const float ur = U[cc][k].re, ui = U[cc][k].im;
                Wg[(2 * k) * 8 + 2 * cc]         = ur;
                Wg[(2 * k + 1) * 8 + 2 * cc]     = -ui;
                Wg[(2 * k) * 8 + 2 * cc + 1]     = ui;
                Wg[(2 * k + 1) * 8 + 2 * cc + 1] = ur;
            }
    }
}

// ---------------------------------------------------------------------------
// TDM descriptor build (CDNA5 ISA). data_size = 8 bytes.
// Tile: tile0 x tile1 x tile2, strides s0/s1 in elements; LDS packed dense.
// ---------------------------------------------------------------------------
__device__ __forceinline__ void make_tdm(unsigned lds_addr,
                                         unsigned long long gaddr,
                                         unsigned tile0, unsigned tile1, unsigned tile2,
                                         unsigned long long s0, unsigned long long s1,
                                         v4u& g0, v8i& g1, v4i& g2, v4i& g3) {
    const unsigned long long td = 1ull << 20;   // generous tensor dims (elements)
    g0.x = 1u;                                  // count=1, user descriptor
    g0.y = lds_addr;                            // LDS byte address
    g0.z = (unsigned)(gaddr & 0xFFFFFFFFull);
    g0.w = (unsigned)((gaddr >> 32) & 0x01FFFFFFull) | (2u << 30);  // type=2
    g1[0] = (int)(3u << 16);                                        // data_size=8B
    g1[1] = (int)((td & 0xFFFFull) << 16);
    g1[2] = (int)(((td >> 16) & 0xFFFFull) | ((td & 0xFFFFull) << 16));
    g1[3] = (int)(((td >> 16) & 0xFFFFull) | ((unsigned long long)tile0 << 16));
    g1[4] = (int)(tile1 | (tile2 << 16));
    g1[5] = (int)(s0 & 0xFFFFFFFFull);
    g1[6] = (int)(((s0 >> 32) & 0xFFFFull) | ((s1 & 0xFFFFull) << 16));
    g1[7] = (int)((s1 >> 16) & 0xFFFFFFFFull);
    g2[0] = (int)(1u << 20); g2[1] = 0; g2[2] = 0; g2[3] = 0;       // tensor_dim2
    g3[0] = 0; g3[1] = 0; g3[2] = 0; g3[3] = 0;
}

__device__ __forceinline__ void tdm_load(unsigned lds_addr, unsigned long long gaddr,
                                         unsigned tile0, unsigned tile1, unsigned tile2,
                                         unsigned long long s0, unsigned long long s1) {
    v4u g0; v8i g1; v4i g2, g3;
    make_tdm(lds_addr, gaddr, tile0, tile1, tile2, s0, s1, g0, g1, g2, g3);
    v8i g4 = { 0, 0, 0, 0, 0, 0, 0, 0 };      // 6-arg form: extra group unused
    __builtin_amdgcn_tensor_load_to_lds(g0, g1, g2, g3, g4, 0);
    __builtin_amdgcn_s_wait_tensorcnt(0);
}

__device__ __forceinline__ void tdm_store(unsigned lds_addr, unsigned long long gaddr,
                                          unsigned tile0, unsigned tile1, unsigned tile2,
                                          unsigned long long s0, unsigned long long s1) {
    v4u g0; v8i g1; v4i g2, g3;
    make_tdm(lds_addr, gaddr, tile0, tile1, tile2, s0, s1, g0, g1, g2, g3);
    v8i g4 = { 0, 0, 0, 0, 0, 0, 0, 0 };
    __builtin_amdgcn_tensor_store_from_lds(g0, g1, g2, g3, g4, 0);
    __builtin_amdgcn_s_wait_tensorcnt(0);
}

// ---------------------------------------------------------------------------
// Apply one fused 4x4-complex gate to an LDS tile via WMMA f32 16x16x4.
// Groups of 4 amplitudes at tile bits (hi, lo); 16 groups per WMMA pair.
// A layout (per ISA): lanes 0-15 supply K0/K1, lanes 16-31 K2/K3 => each lane
// ds_load_b64's exactly one complex amplitude per WMMA.
// B layout assumed to mirror A (v0=row k0, v1=row k0+1; lane half selects k).
// D: lane L holds column n=L&15 of rows m = r + 8*(L>>4).
// ---------------------------------------------------------------------------
__device__ __forceinline__ void apply_gate(float2* tile,
                                           const float* __restrict__ Wg,
                                           const int lo, const int hi,
                                           const int nGroups) {
    const int lane = threadIdx.x & 31;
    const int wid  = threadIdx.x >> 5;
    const int n    = lane & 15;
    const int half = lane >> 4;

    v2f B1 = { 0.f, 0.f }, B2 = { 0.f, 0.f };
    if (n < 8) {
        const int k0 = half << 1;
        B1.x = Wg[(k0 + 0) * 8 + n];
        B1.y = Wg[(k0 + 1) * 8 + n];
        B2.x = Wg[(k0 + 4) * 8 + n];
        B2.y = Wg[(k0 + 5) * 8 + n];
    }

    const unsigned mLo = (1u << lo) - 1u;
    const unsigned mHi = (1u << hi) - 1u;
    const int sLo = 1 << lo, sHi = 1 << hi;

    for (int base = wid * 16; base < nGroups; base += 8 * 16) {
        // group id -> tile index with zeros deposited at bits lo, hi
        unsigned g = (unsigned)(base + n);
        unsigned idx = ((g & ~mLo) << 1) | (g & mLo);
        idx = ((idx & ~mHi) << 1) | (idx & mHi);

        const float2 a1 = tile[idx + half * sLo];          // amp(b1=0, b2=half)
        const float2 a2 = tile[idx + sHi + half * sLo];    // amp(b1=1, b2=half)
        v2f A1 = { a1.x, a1.y };
        v2f A2 = { a2.x, a2.y };

        v8f acc = { 0.f, 0.f, 0.f, 0.f, 0.f, 0.f, 0.f, 0.f };
        acc = __builtin_amdgcn_wmma_f32_16x16x4_f32(false, A1, false, B1,
                                                    (short)0, acc, false, false);
        acc = __builtin_amdgcn_wmma_f32_16x16x4_f32(false, A2, false, B2,
                                                    (short)0, acc, false, false);

        if (n < 8) {
            const int a = n >> 1;          // which of the 4 amplitudes
            const int part = n & 1;        // 0=re, 1=im
            const int aoff = ((a >> 1) ? sHi : 0) + ((a & 1) ? sLo : 0);
#pragma unroll
            for (int r = 0; r < 8; ++r) {
                unsigned go = (unsigned)(base + r + (half << 3));
                unsigned oidx = ((go & ~mLo) << 1) | (go & mLo);
                oidx = ((oidx & ~mHi) << 1) | (oidx & mHi);
                ((float*)&tile[oidx + aoff])[part] = acc[r];
            }
        }
    }
}

// ---------------------------------------------------------------------------
// Pass 1: local bits 0..11, contiguous 4096-complex tiles (32KB LDS).
// Gates (layer order): pairs (8,9)(10,11)(12,13)(14,15)(16,17)(18,19),
// then (8,10)(12,14)(16,18), then (8,12).  bit(q) = 19-q.
// ---------------------------------------------------------------------------
__global__ __launch_bounds__(256) void qcnn_pass1(float* __restrict__ state,
                                                  const float* __restrict__ Wtab) {
    __shared__ float2 tile[4096];
    const unsigned ldsaddr = (unsigned)(uintptr_t)&tile[0];
    const unsigned long long gaddr =
        (unsigned long long)(uintptr_t)state + ((unsigned long long)blockIdx.x << 12) * 8ull;

    if (threadIdx.x < 32u) {
        tdm_load(ldsaddr, gaddr, 4096u, 0u, 0u, 4096ull, 4096ull);
    }
    __syncthreads();

    const int GW[10]  = { 4, 5, 6, 7, 8, 9, 12, 13, 14, 16 };
    const int GLO[10] = { 10, 8, 6, 4, 2, 0, 9, 5, 1, 7 };
    const int GHI[10] = { 11, 9, 7, 5, 3, 1, 11, 7, 3, 11 };
#pragma unroll
    for (int t = 0; t < 10; ++t) {
        apply_gate(tile, Wtab + GW[t] * 64, GLO[t], GHI[t], 1024);
        __syncthreads();
    }

    if (threadIdx.x < 32u) {
        tdm_store(ldsaddr, gaddr, 4096u, 0u, 0u, 4096ull, 4096ull);
    }
}

// ---------------------------------------------------------------------------
// Pass 2: local bits {0..3, 11, 12..19} -> 8192-complex tiles (64KB LDS).
// TDM tile: dim0=16 (bits0..3), dim1=2 stride 2^11 (bit11),
//           dim2=256 stride 2^12 (bits12..19); blockIdx fixes bits 4..10.
// Local coords: bits0..3 -> 0..3, bit11 -> 4, bits12..19 -> 5..12.
// Gates: (0,1)(2,3)(4,5)(6,7), (0,2)(4,6), (0,4), (0,8), (0,16).
// ---------------------------------------------------------------------------
__global__ __launch_bounds__(256) void qcnn_pass2(float* __restrict__ state,
                                                  const float* __restrict__ Wtab) {
    __shared__ float2 tile[8192];
    const unsigned ldsaddr = (unsigned)(uintptr_t)&tile[0];
    const unsigned long long gaddr =
        (unsigned long long)(uintptr_t)state + ((unsigned long long)blockIdx.x << 4) * 8ull;

    if (threadIdx.x < 32u) {
        tdm_load(ldsaddr, gaddr, 16u, 2u, 256u, 2048ull, 4096ull);
    }
    __syncthreads();

    const int GW[9]  = { 0, 1, 2, 3, 10, 11, 15, 17, 18 };
    const int GLO[9] = { 11, 9, 7, 5, 10, 6, 8, 4, 3 };
    const int GHI[9] = { 12, 10, 8, 6, 12, 8, 12, 12, 12 };
#pragma unroll
    for (int t = 0; t < 9; ++t) {
        apply_gate(tile, Wtab + GW[t] * 64, GLO[t], GHI[t], 2048);
        __syncthreads();
    }

    if (threadIdx.x < 32u) {
        tdm_store(ldsaddr, gaddr, 16u, 2u, 256u, 2048ull, 4096ull);
    }
}

// ---------------------------------------------------------------------------
// Reduction: out = P(q0=0) = sum over first 2^19 complex of |amp|^2
//          = sum of squares of first 2^20 floats.
// ---------------------------------------------------------------------------
__global__ __launch_bounds__(256) void qcnn_reduce1(const float* __restrict__ state,
                                                    float* __restrict__ partials) {
    __shared__ float sm[256];
    float acc = 0.f;
    for (unsigned i = blockIdx.x * 256u + threadIdx.x; i < (1u << 20);
         i += gridDim.x * 256u) {
        const float v = state[i];
        acc += v * v;
    }
    sm[threadIdx.x] = acc;
    __syncthreads();
    for (int s = 128; s > 0; s >>= 1) {
        if ((int)threadIdx.x < s) sm[threadIdx.x] += sm[threadIdx.x + s];
        __syncthreads();
    }
    if (threadIdx.x == 0) partials[blockIdx.x] = sm[0];
}

__global__ __launch_bounds__(128) void qcnn_reduce2(const float* __restrict__ partials,
                                                    float* __restrict__ out) {
    __shared__ float sm[128];
    sm[threadIdx.x] = partials[threadIdx.x];
    __syncthreads();
    for (int s = 64; s > 0; s >>= 1) {
        if ((int)threadIdx.x < s) sm[threadIdx.x] += sm[threadIdx.x + s];
        __syncthreads();
    }
    if (threadIdx.x == 0) out[0] = sm[0];
}

// ---------------------------------------------------------------------------
extern "C" void kernel_launch(void* const* d_in, const int* in_sizes, int n_in,
                              void* d_out, int out_size, void* d_ws, size_t ws_size,
                              hipStream_t stream) {
    (void)in_sizes; (void)n_in; (void)out_size; (void)ws_size;
    const float* params = (const float*)d_in[0];          // 57 floats
    float* out = (float*)d_out;                           // 1 float

    // Workspace layout: state (8MB) | W tables (19*64 f32) | partials (128 f32)
    float* state    = (float*)d_ws;
    float* Wtab     = (float*)((char*)d_ws + (8u << 20));
    float* partials = Wtab + 19 * 64;

    qcnn_init   <<<64, 256, 0, stream>>>(params, state, Wtab);
    qcnn_pass1  <<<256, 256, 0, stream>>>(state, Wtab);
    qcnn_pass2  <<<128, 256, 0, stream>>>(state, Wtab);
    qcnn_reduce1<<<128, 256, 0, stream>>>(state, partials);
    qcnn_reduce2<<<1, 128, 0, stream>>>(partials, out);
}